// Attention_44976897523785
// MI455X (gfx1250) — compile-verified
//
#include <hip/hip_runtime.h>
#include <hip/hip_bf16.h>

typedef __attribute__((ext_vector_type(2))) float v2f;
typedef __attribute__((ext_vector_type(8))) float v8f;

#define BB 32
#define SS 2048
#define HH 1024
#define SC 128              // S rows per fused block (one partial per chunk)
#define GG (SS / SC)        // 16 chunks per batch
#define TS 32               // rows per LDS-staged sub-tile
#define NSUB (SC / TS)      // 4 sub-tiles per chunk
#define NT2 512             // threads per fused block (16 waves)
#define NT 256              // threads for K3

// ---------------------------------------------------------------------------
// K1: q = hidden @ W^T + bias   via V_WMMA_F32_16X16X4_F32 (verified lowering).
// One wave per 16x16 output tile, K=1024 chained through the accumulator.
// ---------------------------------------------------------------------------
__global__ __launch_bounds__(32) void qgemm_wmma(const float* __restrict__ hidden,
                                                 const float* __restrict__ W,
                                                 const float* __restrict__ bias,
                                                 float* __restrict__ qout) {
  const int tile = blockIdx.x;     // 0..127
  const int tm   = tile & 1;       // 2 M-tiles
  const int tn   = tile >> 1;      // 64 N-tiles
  const int lane = threadIdx.x;    // 0..31
  const int half = lane >> 4;      // 0: K pair {0,1}; 1: K pair {2,3}
  const int l16  = lane & 15;
  const int m0 = tm * 16;
  const int n0 = tn * 16;

  const float* arow = hidden + (size_t)(m0 + l16) * HH + 2 * half; // A[m][k]
  const float* brow = W      + (size_t)(n0 + l16) * HH + 2 * half; // B[k][n]=W[n][k]

  v8f acc = {};
#pragma unroll 8
  for (int k0 = 0; k0 < HH; k0 += 4) {
    const float2 av = *(const float2*)(arow + k0);
    const float2 bv = *(const float2*)(brow + k0);
    v2f a, b;
    a[0] = av.x; a[1] = av.y;
    b[0] = bv.x; b[1] = bv.y;
    acc = __builtin_amdgcn_wmma_f32_16x16x4_f32(false, a, false, b,
                                                (short)0, acc, false, false);
  }

  const float bsv = bias[n0 + l16];
#pragma unroll
  for (int r = 0; r < 8; ++r) {
    const int m = m0 + r + (half ? 8 : 0);
    qout[(size_t)m * HH + n0 + l16] = acc[r] + bsv;
  }
}

// ---------------------------------------------------------------------------
// Async DMA of one 32-row x 1024-col f32 sub-tile (128KB) into LDS.
// 512 threads x 16 b128 transfers each; ASYNCcnt += 16 per wave.
// ---------------------------------------------------------------------------
__device__ __forceinline__ void issue_subtile_async(const float* __restrict__ gbase,
                                                    float* __restrict__ lbuf,
                                                    int tid) {
  const float4* g4 = (const float4*)gbase;
  float4* l4 = (float4*)lbuf;
  for (int f = tid; f < TS * HH / 4; f += NT2) {
    const unsigned loff = (unsigned)(uintptr_t)(l4 + f);   // LDS byte offset
    const float4* ga = g4 + f;
    asm volatile("global_load_async_to_lds_b128 %0, %1, off"
                 :: "v"(loff), "v"(ga) : "memory");
  }
}

// ---------------------------------------------------------------------------
// K2: flash-style fused pass. grid = (GG, BB), 512 threads.
// enc chunk streamed through LDS exactly once (double-buffered async DMA):
//   per sub-tile: energies from LDS, online max/sum-exp update (rescale
//   running context), weighted accumulation from LDS.
// ---------------------------------------------------------------------------
__global__ __launch_bounds__(NT2) void attn_partial(const float* __restrict__ enc,
                                                    const float* __restrict__ q,
                                                    float* __restrict__ energy_out,
                                                    float* __restrict__ mpart,
                                                    float* __restrict__ zpart,
                                                    float* __restrict__ ctxpart) {
  extern __shared__ float dynbuf[];                 // 2 x 32K floats (256KB)

  __shared__ float q_sh[HH];
  __shared__ float e_sub[TS];
  __shared__ float w_sh[TS];
  __shared__ float m_run_sh, z_run_sh, rescale_sh;

  const int b = blockIdx.y, g = blockIdx.x;
  const int tid = threadIdx.x, lane = tid & 31, wave = tid >> 5;
  const int s0 = g * SC;
  const float* encb = enc + ((size_t)b * SS + s0) * HH;

  // stage q[b,:] (plain loads; does not perturb ASYNCcnt accounting)
  for (int i = tid; i < HH; i += NT2) q_sh[i] = q[b * HH + i];
  if (tid == 0) { m_run_sh = -1e30f; z_run_sh = 0.f; }

  // prologue: kick off sub-tile 0 into buffer parity 0
  issue_subtile_async(encb, dynbuf, tid);

  float acc0 = 0.f, acc1 = 0.f;                     // running context, 2 cols/thread
  const float4* q4 = (const float4*)q_sh;

  for (int t = 0; t < NSUB; ++t) {
    // prefetch next sub-tile into the other buffer, then wait for current.
    if (t + 1 < NSUB) {
      issue_subtile_async(encb + (size_t)(t + 1) * TS * HH,
                          dynbuf + (size_t)((t + 1) & 1) * TS * HH, tid);
      // 32 outstanding per wave; in-order completion => <=16 means buf[t] done
      asm volatile("s_wait_asynccnt 0x10" ::: "memory");
    } else {
      asm volatile("s_wait_asynccnt 0x0" ::: "memory");
    }
    __syncthreads();                                // buf[t] visible to all waves

    const float* tb = dynbuf + (size_t)(t & 1) * TS * HH;
    const float4* tb4 = (const float4*)tb;

    // ---- energies: 16 waves x 2 rows, float4 LDS reads ----
    for (int r = wave; r < TS; r += NT2 / 32) {
      const float4* row4 = tb4 + (size_t)r * (HH / 4);
      float acc = 0.f;
#pragma unroll
      for (int i = lane; i < HH / 4; i += 32) {
        const float4 v = row4[i];
        const float4 w = q4[i];
        acc = fmaf(v.x, w.x, acc);
        acc = fmaf(v.y, w.y, acc);
        acc = fmaf(v.z, w.z, acc);
        acc = fmaf(v.w, w.w, acc);
      }
#pragma unroll
      for (int off = 16; off > 0; off >>= 1) acc += __shfl_down(acc, off, 32);
      if (lane == 0) e_sub[r] = acc;
    }
    __syncthreads();

    // raw energies out (finalized in K3)
    if (tid < TS) energy_out[(size_t)b * SS + s0 + t * TS + tid] = e_sub[tid];

    // ---- online softmax update (wave 0) ----
    if (wave == 0) {
      const float ev = e_sub[lane];
      float mx = ev;
#pragma unroll
      for (int off = 16; off > 0; off >>= 1) mx = fmaxf(mx, __shfl_xor(mx, off, 32));
      const float newm = fmaxf(m_run_sh, mx);
      const float wv = __expf(ev - newm);
      float sum = wv;
#pragma unroll
      for (int off = 16; off > 0; off >>= 1) sum += __shfl_xor(sum, off, 32);
      w_sh[lane] = wv;
      if (lane == 0) {
        const float rs = __expf(m_run_sh - newm);
        rescale_sh = rs;
        z_run_sh = z_run_sh * rs + sum;
        m_run_sh = newm;
      }
    }
    __syncthreads();

    // ---- rescale running context, accumulate this sub-tile from LDS ----
    const float rs = rescale_sh;
    acc0 *= rs;
    acc1 *= rs;
    for (int r = 0; r < TS; ++r) {
      const float wr = w_sh[r];
      acc0 = fmaf(wr, tb[(size_t)r * HH + tid        ], acc0);
      acc1 = fmaf(wr, tb[(size_t)r * HH + tid + NT2  ], acc1);
    }
    __syncthreads();                                // done reading buf[t]
  }

  // partials for K3
  float* cp = ctxpart + (size_t)(b * GG + g) * HH;
  cp[tid]       = acc0;
  cp[tid + NT2] = acc1;
  if (tid == 0) {
    mpart[b * GG + g] = m_run_sh;
    zpart[b * GG + g] = z_run_sh;
  }
}

// ---------------------------------------------------------------------------
// K3: combine partials, normalize attention in place, reduce partial contexts.
// ---------------------------------------------------------------------------
__global__ __launch_bounds__(NT) void attn_final(const float* __restrict__ mpart,
                                                 const float* __restrict__ zpart,
                                                 const float* __restrict__ ctxpart,
                                                 float* __restrict__ ctx_out,
                                                 float* __restrict__ attn_out) {
  __shared__ float scale[GG];
  __shared__ float sM, sInvZ;
  const int b = blockIdx.x, tid = threadIdx.x;

  if (tid == 0) {
    float M = -1e30f;
    for (int g = 0; g < GG; ++g) M = fmaxf(M, mpart[b * GG + g]);
    float Z = 0.f;
    for (int g = 0; g < GG; ++g) {
      const float sc = __expf(mpart[b * GG + g] - M);
      scale[g] = sc;
      Z += sc * zpart[b * GG + g];
    }
    sM = M;
    sInvZ = 1.f / Z;
  }
  __syncthreads();
  const float M = sM, invZ = sInvZ;

  for (int i = tid; i < SS; i += NT) {
    const size_t idx = (size_t)b * SS + i;
    attn_out[idx] = __expf(attn_out[idx] - M) * invZ;
  }

  float a0 = 0.f, a1 = 0.f, a2 = 0.f, a3 = 0.f;
  for (int g = 0; g < GG; ++g) {
    const float* cp = ctxpart + (size_t)(b * GG + g) * HH;
    const float sc = scale[g];
    a0 = fmaf(sc, cp[tid        ], a0);
    a1 = fmaf(sc, cp[tid + 256  ], a1);
    a2 = fmaf(sc, cp[tid + 512  ], a2);
    a3 = fmaf(sc, cp[tid + 768  ], a3);
  }
  float* co = ctx_out + (size_t)b * HH;
  co[tid]       = a0 * invZ;
  co[tid + 256] = a1 * invZ;
  co[tid + 512] = a2 * invZ;
  co[tid + 768] = a3 * invZ;
}

// ---------------------------------------------------------------------------
extern "C" void kernel_launch(void* const* d_in, const int* in_sizes, int n_in,
                              void* d_out, int out_size, void* d_ws, size_t ws_size,
                              hipStream_t stream) {
  const float* hidden = (const float*)d_in[0];  // [B,H]
  const float* enc    = (const float*)d_in[1];  // [B,S,H]
  const float* W      = (const float*)d_in[2];  // [H,H] (out,in)
  const float* bias   = (const float*)d_in[3];  // [H]

  float* ctx_out  = (float*)d_out;              // [B,H]   (first output)
  float* attn_out = (float*)d_out + BB * HH;    // [B,S]   (second output)

  // workspace layout (floats): q | mpart | zpart | ctxpart  (~2.23 MB total)
  float* ws      = (float*)d_ws;
  float* q       = ws;                       // B*H      = 32768
  float* mpart   = q + BB * HH;              // B*GG     = 512
  float* zpart   = mpart + BB * GG;          // B*GG     = 512
  float* ctxpart = zpart + BB * GG;          // B*GG*H   = 524288

  const size_t dynls = (size_t)2 * TS * HH * sizeof(float);   // 256KB double buffer
  (void)hipFuncSetAttribute((const void*)attn_partial,
                            hipFuncAttributeMaxDynamicSharedMemorySize,
                            (int)dynls);

  qgemm_wmma<<<128, 32, 0, stream>>>(hidden, W, bias, q);

  dim3 grid2(GG, BB);
  attn_partial<<<grid2, NT2, dynls, stream>>>(enc, q, attn_out, mpart, zpart, ctxpart);

  attn_final<<<BB, NT, 0, stream>>>(mpart, zpart, ctxpart, ctx_out, attn_out);
}